// FaceDetector_70901320123010
// MI455X (gfx1250) — compile-verified
//
#include <hip/hip_runtime.h>
#include <stdint.h>

#define BB   64
#define NN   21504
#define KK   1024
#define NP   32768          // NN padded to pow2 for bitonic sort
#define DETS_FLOATS  (BB * KK * 5)   // 327680
#define KEEP_OFFSET  DETS_FLOATS

typedef __attribute__((ext_vector_type(2))) float v2f;
typedef __attribute__((ext_vector_type(8))) float v8f;

static __device__ __forceinline__ unsigned ballot32(bool p) {
    return __builtin_amdgcn_ballot_w32(p);
}

// ---------------------------------------------------------------------------
// Kernel 1: per-image top-K (sorted desc, stable) + box decode.
// One 1024-thread block per image; 256KB dynamic LDS bitonic sort of 64b keys.
// Writes UNMASKED dets [b][k][5] = (x1,y1,x2,y2,score) into d_out.
// ---------------------------------------------------------------------------
__global__ void __launch_bounds__(1024)
fd_topk_decode(const float* __restrict__ locs,
               const float* __restrict__ confs,
               const float* __restrict__ priors,
               float* __restrict__ out) {
    extern __shared__ unsigned long long keys[];   // NP entries = 256KB
    const int b   = blockIdx.x;
    const int tid = threadIdx.x;

    // ---- build sortable keys: (monotone(c1-c0) << 32) | ~idx  (stable ties)
    for (int idx = tid; idx < NP; idx += 1024) {
        unsigned long long key = 0ull;
        if (idx < NN) {
            const float* cp = confs + ((size_t)b * NN + idx) * 2;
            __builtin_prefetch(cp + 2 * 1024, 0, 0);   // linear stream hint -> global_prefetch_b8
            float diff = cp[1] - cp[0];
            unsigned bits = __float_as_uint(diff);
            unsigned s = bits ^ ((bits >> 31) ? 0xFFFFFFFFu : 0x80000000u);
            key = ((unsigned long long)s << 32) | (unsigned)(~idx);
        }
        keys[idx] = key;
    }
    __syncthreads();

    // ---- bitonic sort, descending
    for (unsigned size = 2; size <= NP; size <<= 1) {
        for (unsigned stride = size >> 1; stride > 0; stride >>= 1) {
            for (unsigned i = tid; i < NP / 2; i += 1024) {
                unsigned pos = 2u * i - (i & (stride - 1));
                bool desc = ((pos & size) == 0);
                unsigned long long a = keys[pos];
                unsigned long long c = keys[pos + stride];
                bool sw = desc ? (a < c) : (a > c);
                if (sw) { keys[pos] = c; keys[pos + stride] = a; }
            }
            __syncthreads();
        }
    }

    // ---- decode top-K and emit unmasked dets
    {
        const int k = tid;                       // 1024 threads == K
        unsigned long long key = keys[k];
        int idx = (int)(~(unsigned)(key & 0xFFFFFFFFull));
        const float* cp = confs + ((size_t)b * NN + idx) * 2;
        float score = 1.0f / (1.0f + __expf(cp[0] - cp[1]));

        const float4 l = *(const float4*)(locs + ((size_t)b * NN + idx) * 4);
        const float4 p = *(const float4*)(priors + (size_t)idx * 4);
        float cx = p.x + l.x * 0.1f * p.z;
        float cy = p.y + l.y * 0.1f * p.w;
        float w  = p.z * __expf(l.z * 0.2f);
        float h  = p.w * __expf(l.w * 0.2f);
        float x1 = cx - w * 0.5f;
        float y1 = cy - h * 0.5f;

        float* o = out + ((size_t)b * KK + k) * 5;
        o[0] = x1; o[1] = y1; o[2] = x1 + w; o[3] = y1 + h; o[4] = score;
    }
}

// ---------------------------------------------------------------------------
// Kernel 2: per-image greedy NMS via bitmask matrix.
//   - async-copy dets tile into LDS (global_load_async_to_lds_b128)
//   - 16x16 tiles: union base (area_i + area_j) via v_wmma_f32_16x16x4_f32,
//     intersection elementwise in the WMMA C/D lane layout, ballot -> bits
//   - division-free threshold: iou > t  <=>  inter > t * max(union, eps)
//   - one wave32 holds the 1024-bit removed mask (1 u32/lane) for the
//     sequential greedy sweep
// ---------------------------------------------------------------------------
__global__ void __launch_bounds__(1024)
fd_nms(float* __restrict__ out) {
    extern __shared__ char smem[];
    float*    raw    = (float*)smem;                         // 5120 f = 20480B
    float4*   boxesL = (float4*)(smem + 20480);              // 1024 f4 = 16384B
    float*    areaL  = (float*)(smem + 36864);               // 1024 f  =  4096B
    unsigned* supmat = (unsigned*)(smem + 40960);            // 1024*32 u32 = 128KB
    unsigned* validw = (unsigned*)(smem + 40960 + 131072);   // 32 u32
    unsigned* keepw  = validw + 32;                          // 32 u32

    const int b    = blockIdx.x;
    const int tid  = threadIdx.x;
    const int lane = tid & 31;
    const int wave = tid >> 5;

    // ---- stage 20480B of dets for this image into LDS asynchronously
    {
        const char* src = (const char*)out + (size_t)b * (KK * 5 * 4);
        unsigned lds_base = (unsigned)(uintptr_t)(void*)raw;
        unsigned dst0 = lds_base + (unsigned)tid * 16u;
        const char* s0 = src + (size_t)tid * 16;
        asm volatile("global_load_async_to_lds_b128 %0, %1, off"
                     :: "v"(dst0), "v"(s0) : "memory");
        if (tid < 256) {
            unsigned dst1 = lds_base + (unsigned)(1024 + tid) * 16u;
            const char* s1 = src + (size_t)(1024 + tid) * 16;
            asm volatile("global_load_async_to_lds_b128 %0, %1, off"
                         :: "v"(dst1), "v"(s1) : "memory");
        }
        asm volatile("s_wait_asynccnt 0x0" ::: "memory");
    }
    __syncthreads();

    // ---- unpack boxes / areas, zero suppression matrix, build valid bitmask
    {
        float x1 = raw[tid * 5 + 0], y1 = raw[tid * 5 + 1];
        float x2 = raw[tid * 5 + 2], y2 = raw[tid * 5 + 3];
        float sc = raw[tid * 5 + 4];
        boxesL[tid] = make_float4(x1, y1, x2, y2);
        areaL[tid]  = fmaxf(x2 - x1, 0.0f) * fmaxf(y2 - y1, 0.0f);
        unsigned bal = ballot32(sc > 0.5f);
        if (lane == 0) validw[wave] = bal;
        for (int w2 = tid; w2 < KK * 32; w2 += 1024) supmat[w2] = 0u;
    }
    __syncthreads();

    // ---- suppression bit-matrix over 16x16 tiles (upper triangle)
    for (int t = wave; t < 64 * 64; t += 32) {
        const int ti = t >> 6, tj = t & 63;
        if (tj < ti) continue;
        const int i0 = ti * 16, j0 = tj * 16;

        // A (16x4): col0 = area_i, col1 = 1     -> lanes 0-15 hold K=0,1
        // B (4x16): row0 = 1, row1 = area_j     -> lanes 0-15 hold K=0,1
        bool lo = (lane < 16);
        v2f a = { lo ? areaL[i0 + lane] : 0.0f, lo ? 1.0f : 0.0f };
        v2f bf = { lo ? 1.0f : 0.0f, lo ? areaL[j0 + lane] : 0.0f };
        v8f c = {};
        v8f d = __builtin_amdgcn_wmma_f32_16x16x4_f32(
            false, a, false, bf, (short)0, c, false, false);
        // d[r] = area[i0+M] + area[j0+N], M = r + (lane>=16)*8, N = lane&15

        const int moff = (lane >= 16) ? 8 : 0;
        const int j    = j0 + (lane & 15);
        const float4 bj = boxesL[j];
        const int wsel  = j0 >> 5;
        const int wsh   = j0 & 16;
        const int rowadd = lane ? 8 : 0;   // lane0 -> rows M, lane1 -> rows M+8
#pragma unroll
        for (int r = 0; r < 8; ++r) {
            const int i = i0 + r + moff;
            const float4 bi = boxesL[i];
            float iw = fmaxf(fminf(bi.z, bj.z) - fmaxf(bi.x, bj.x), 0.0f);
            float ih = fmaxf(fminf(bi.w, bj.w) - fmaxf(bi.y, bj.y), 0.0f);
            float inter = iw * ih;
            float uni   = fmaxf(d[r] - inter, 1e-9f);     // WMMA union base
            // iou > 0.4  <=>  inter > 0.4 * uni   (uni > 0): no divide
            unsigned bal = ballot32((inter > 0.4f * uni) && (j > i));
            unsigned bits = lane ? (bal >> 16) : (bal & 0xFFFFu);
            if (lane < 2 && bits)
                atomicOr(&supmat[(i0 + r + rowadd) * 32 + wsel], bits << wsh);
        }
    }
    __syncthreads();

    // ---- sequential greedy sweep: wave 0 holds removed mask (1 u32 / lane)
    if (tid < 32) {
        unsigned removed = 0u;
        unsigned vw = validw[lane];
        for (int i = 0; i < KK; ++i) {
            int ws = i >> 5;
            unsigned bit = 1u << (i & 31);
            unsigned rw = __shfl(removed, ws, 32);
            unsigned vv = __shfl(vw, ws, 32);
            bool keep_i = (vv & bit) && !(rw & bit);
            if (keep_i) removed |= supmat[i * 32 + lane];
        }
        keepw[lane] = vw & ~removed;
    }
    __syncthreads();

    // ---- masked outputs
    {
        const int k = tid;
        float kf = ((keepw[k >> 5] >> (k & 31)) & 1u) ? 1.0f : 0.0f;
        float* o = out + ((size_t)b * KK + k) * 5;
#pragma unroll
        for (int cc = 0; cc < 5; ++cc) o[cc] = raw[k * 5 + cc] * kf;
        out[(size_t)KEEP_OFFSET + (size_t)b * KK + k] = kf;
    }
}

extern "C" void kernel_launch(void* const* d_in, const int* in_sizes, int n_in,
                              void* d_out, int out_size, void* d_ws, size_t ws_size,
                              hipStream_t stream) {
    (void)in_sizes; (void)n_in; (void)d_ws; (void)ws_size; (void)out_size;
    const float* locs   = (const float*)d_in[0];
    const float* confs  = (const float*)d_in[1];
    const float* priors = (const float*)d_in[2];
    float* out = (float*)d_out;

    const int LDS_TOPK = NP * 8;                         // 256 KB
    const int LDS_NMS  = 40960 + 131072 + 64 * 4;        // ~172 KB
    (void)hipFuncSetAttribute((const void*)fd_topk_decode,
                              hipFuncAttributeMaxDynamicSharedMemorySize, LDS_TOPK);
    (void)hipFuncSetAttribute((const void*)fd_nms,
                              hipFuncAttributeMaxDynamicSharedMemorySize, LDS_NMS);

    fd_topk_decode<<<BB, 1024, LDS_TOPK, stream>>>(locs, confs, priors, out);
    fd_nms<<<BB, 1024, LDS_NMS, stream>>>(out);
}